// YOLOV3Loss_89988154786569
// MI455X (gfx1250) — compile-verified
//
#include <hip/hip_runtime.h>
#include <math.h>

// ---------------- problem constants (fixed by reference setup_inputs) -------
#define BATCH        32
#define NGT          32
#define NUM_CLASSES  80
#define IOU_THR      0.7f
#define LAMBDA_OBJ   5.0f

// per-scale box counts (3 anchors * D*D, D = 13,26,52)
#define NB0  507
#define NB1  2028
#define NB2  8112
#define SB0  0
#define SB1  (BATCH*NB0)              // 16224
#define SB2  (SB1 + BATCH*NB1)        // 81120
#define TOTBOX (SB2 + BATCH*NB2)      // 340704

#define NREC (3*BATCH*NGT)            // 3072

// workspace layout (bytes)
#define FLAG_WORDS (TOTBOX/4)         // 85176 u32 words (TOTBOX % 4 == 0)
#define REC_OFF   344064              // aligned past flags
#define REC_STRIDE 8                  // floats per record
#define PART_OFF  (REC_OFF + NREC*REC_STRIDE*4)   // 442368

// partial-sum slots (one per block, fixed order -> deterministic)
#define BLK0 ((BATCH*NB0 + 255)/256)  // 64
#define BLK1 ((BATCH*NB1 + 255)/256)  // 254
#define BLK2 ((BATCH*NB2 + 255)/256)  // 1014
#define BLKT ((NREC + 255)/256)       // 12
#define SLOT0 0
#define SLOT1 (SLOT0 + BLK0)
#define SLOT2 (SLOT1 + BLK1)
#define SLOTT (SLOT2 + BLK2)
#define NPART (SLOTT + BLKT)          // 1344

typedef __attribute__((ext_vector_type(2))) float v2f;
typedef __attribute__((ext_vector_type(8))) float v8f;

// fast sigmoid: v_exp_f32 + v_rcp_f32 (1 ULP; feeds squared loss terms)
__device__ __forceinline__ float sigf(float x) {
    return __builtin_amdgcn_rcpf(1.0f + __expf(-x));
}

// Wave32 reduction through the matrix pipe: A(16x4) holds lane values in K0/K2
// (VGPR0) with K1/K3 = 0 (VGPR1); B = ones(4x16).  D[m][n] = lane_m + lane_{m+16}
// in f32.  Summing a lane's 8 C-VGPRs gives a column partial; adding the
// partner half-wave's partial (shfl 16) yields the full wave sum.
__device__ __forceinline__ float wave_sum_wmma(float v) {
    v2f a; a[0] = v;    a[1] = 0.0f;
    v2f b; b[0] = 1.0f; b[1] = 1.0f;
    v8f c = {};
    c = __builtin_amdgcn_wmma_f32_16x16x4_f32(false, a, false, b,
                                              (short)0, c, false, false);
    float s = c[0] + c[1] + c[2] + c[3] + c[4] + c[5] + c[6] + c[7];
    s += __shfl_down(s, 16, 32);
    return s;   // lanes 0..15 hold the full wave sum
}

// block (256 = 8 waves) reduction; every block writes exactly one slot
__device__ __forceinline__ void block_accum(float v, float* slot) {
    __shared__ float wred[8];
    float t = wave_sum_wmma(v);
    int lane = threadIdx.x & 31, wid = threadIdx.x >> 5;
    if (lane == 0) wred[wid] = t;
    __syncthreads();
    if (threadIdx.x == 0) {
        float s = 0.0f;
        #pragma unroll
        for (int i = 0; i < 8; ++i) s += wred[i];
        *slot = s;
    }
}

// ------------------------------------------------------------------ kernels
__global__ void k_zero(unsigned int* flags_w) {
    int i = blockIdx.x * blockDim.x + threadIdx.x;
    if (i < FLAG_WORDS) flags_w[i] = 0u;
}

// one thread per (scale, batch, gt): anchor assignment + targets + flag scatter
__global__ void k_assign(const float* __restrict__ gt_boxes,
                         const float* __restrict__ anchors,
                         unsigned char* __restrict__ flags,
                         float* __restrict__ rec) {
    int tid = blockIdx.x * blockDim.x + threadIdx.x;
    if (tid >= NREC) return;
    int s = tid / (BATCH * NGT);
    int r = tid - s * (BATCH * NGT);
    int b = r / NGT, g = r - b * NGT;

    int D = 13 << s;                         // 13, 26, 52
    int nb = 3 * D * D;
    int sbase = (s == 0) ? SB0 : (s == 1 ? SB1 : SB2);

    const float* gt = gt_boxes + 4 * (b * NGT + g);
    float cx = gt[0], cy = gt[1], gw = gt[2], gh = gt[3];

    int cell_x = (int)floorf(cx * (float)D);
    int cell_y = (int)floorf(cy * (float)D);
    cell_x = min(max(cell_x, 0), D - 1);
    cell_y = min(max(cell_y, 0), D - 1);
    float dx = cx - (float)cell_x / (float)D;
    float dy = cy - (float)cell_y / (float)D;

    // best anchor of this scale by wh-IoU (strict > keeps first max, like argmax)
    int bp = 0; float bestr = -1.0f;
    #pragma unroll
    for (int a = 0; a < 3; ++a) {
        float aw = anchors[(s * 3 + a) * 2 + 0];
        float ah = anchors[(s * 3 + a) * 2 + 1];
        float inter = fminf(gw, aw) * fminf(gh, ah);
        float un = gw * gh + aw * ah - inter;
        float rr = inter / (un + 1e-9f);
        if (rr > bestr) { bestr = rr; bp = a; }
    }
    int bb = bp * D * D + cell_y * D + cell_x;

    flags[sbase + b * nb + bb] = 1;          // benign same-value race

    float aw = anchors[(s * 3 + bp) * 2 + 0];
    float ah = anchors[(s * 3 + bp) * 2 + 1];
    float* rc = rec + tid * REC_STRIDE;
    ((int*)rc)[0] = bb;
    ((int*)rc)[1] = bp;
    rc[2] = dx;
    rc[3] = dy;
    rc[4] = __logf(gw) - __logf(aw);
    rc[5] = __logf(gh) - __logf(ah);
}

// one thread per predicted box of one scale: noobj term.
// Division-free threshold:  iou > THR  <=>  (1+THR)*inter > THR*(pa + ga + eps)
__global__ void __launch_bounds__(256)
k_noobj(const float* __restrict__ pred,
        const float* __restrict__ anc6,            // anchors + s*6
        const float* __restrict__ gt_boxes,
        const unsigned char* __restrict__ flags_s, // flags + scale base
        int D, int nb, int total,
        float* __restrict__ part_slot_base) {
    __shared__ float gx1[BATCH * NGT], gy1[BATCH * NGT], gx2[BATCH * NGT],
                     gy2[BATCH * NGT], g07[BATCH * NGT];
    for (int k = threadIdx.x; k < BATCH * NGT; k += blockDim.x) {
        float cx = gt_boxes[4 * k], cy = gt_boxes[4 * k + 1];
        float w = gt_boxes[4 * k + 2], h = gt_boxes[4 * k + 3];
        gx1[k] = cx - 0.5f * w; gy1[k] = cy - 0.5f * h;
        gx2[k] = cx + 0.5f * w; gy2[k] = cy + 0.5f * h;
        g07[k] = IOU_THR * (w * h + 1e-9f);
    }
    __syncthreads();

    int i = blockIdx.x * blockDim.x + threadIdx.x;
    float val = 0.0f;
    if (i < total) {
        int hw = D * D;
        int b = i / nb;
        int r = i - b * nb;
        int a = r / hw;
        int yx = r - a * hw;
        int y = yx / D, x = yx - y * D;

        const float* pb = pred + ((size_t)b * 255 + (size_t)a * 85) * hw + yx;
        float p0 = pb[0], p1 = pb[hw], p2 = pb[2 * hw], p3 = pb[3 * hw], p4 = pb[4 * hw];
        float sobj = sigf(p4);

        if (!flags_s[i]) {                   // flagged -> pos==1.0, excluded
            float aw = anc6[2 * a], ah = anc6[2 * a + 1];
            float pcx = (sigf(p0) + (float)x) / (float)D;
            float pcy = (sigf(p1) + (float)y) / (float)D;
            float pw = aw * __expf(p2), ph = ah * __expf(p3);
            float px1 = pcx - 0.5f * pw, py1 = pcy - 0.5f * ph;
            float px2 = pcx + 0.5f * pw, py2 = pcy + 0.5f * ph;
            float pa07 = IOU_THR * (pw * ph);

            // best_m = max_g [ (1+THR)*inter - THR*(ga+eps) ];  hit <=> best_m > pa07
            float best_m = -3.0e38f;
            int gb = b * NGT;
            #pragma unroll 4
            for (int g = 0; g < NGT; ++g) {
                float ix1 = fmaxf(px1, gx1[gb + g]);
                float iy1 = fmaxf(py1, gy1[gb + g]);
                float ix2 = fminf(px2, gx2[gb + g]);
                float iy2 = fminf(py2, gy2[gb + g]);
                float iw = fmaxf(0.0f, ix2 - ix1);
                float ih = fmaxf(0.0f, iy2 - iy1);
                float m = fmaf((1.0f + IOU_THR), iw * ih, -g07[gb + g]);
                best_m = fmaxf(best_m, m);
            }
            if (best_m <= pa07) val = sobj * sobj;   // pos == 0 -> noobj
        }
    }
    block_accum(val, part_slot_base + blockIdx.x);
}

// one thread per (scale,batch,gt): coord + obj + cls at best box
__global__ void k_targets(const float* __restrict__ p0d,
                          const float* __restrict__ p1d,
                          const float* __restrict__ p2d,
                          const float* __restrict__ gt_boxes,
                          const int* __restrict__ gt_classes,
                          const float* __restrict__ anchors,
                          const float* __restrict__ rec,
                          float* __restrict__ part_slot_base) {
    int tid = blockIdx.x * blockDim.x + threadIdx.x;
    float loss = 0.0f;
    if (tid < NREC) {
        int s = tid / (BATCH * NGT);
        int r = tid - s * (BATCH * NGT);
        int b = r / NGT, g = r - b * NGT;

        int D = 13 << s, hw = D * D;
        const float* pred = (s == 0) ? p0d : (s == 1 ? p1d : p2d);
        const float* rc = rec + tid * REC_STRIDE;
        int bb = ((const int*)rc)[0];
        int bp = ((const int*)rc)[1];
        float dx = rc[2], dy = rc[3], tlw = rc[4], tlh = rc[5];

        int a = bb / hw;
        int yx = bb - a * hw;
        int y = yx / D, x = yx - y * D;
        const float* pb = pred + ((size_t)b * 255 + (size_t)a * 85) * hw + yx;

        float c0 = pb[0], c1 = pb[hw], c2 = pb[2 * hw], c3 = pb[3 * hw], c4 = pb[4 * hw];
        float sx = sigf(c0), sy = sigf(c1);

        // coord (LAMBDA_COORD = 1)
        float d0 = sx - dx, d1 = sy - dy, d2 = c2 - tlw, d3 = c3 - tlh;
        loss += d0 * d0 + d1 * d1 + d2 * d2 + d3 * d3;

        // t_iou = IoU(gt_g, decoded best box)   (precise divide; 3072 total)
        float aw = anchors[(s * 3 + bp) * 2 + 0];   // a == bp by construction
        float ah = anchors[(s * 3 + bp) * 2 + 1];
        float pcx = (sx + (float)x) / (float)D;
        float pcy = (sy + (float)y) / (float)D;
        float pw = aw * __expf(c2), ph = ah * __expf(c3);
        const float* gt = gt_boxes + 4 * (b * NGT + g);
        float gcx = gt[0], gcy = gt[1], gw = gt[2], gh = gt[3];
        float ix1 = fmaxf(pcx - 0.5f * pw, gcx - 0.5f * gw);
        float iy1 = fmaxf(pcy - 0.5f * ph, gcy - 0.5f * gh);
        float ix2 = fminf(pcx + 0.5f * pw, gcx + 0.5f * gw);
        float iy2 = fminf(pcy + 0.5f * ph, gcy + 0.5f * gh);
        float inter = fmaxf(0.0f, ix2 - ix1) * fmaxf(0.0f, iy2 - iy1);
        float tiou = inter / (pw * ph + gw * gh - inter + 1e-9f);

        float dobj = sigf(c4) - tiou;
        loss += LAMBDA_OBJ * dobj * dobj;

        int cls = gt_classes[b * NGT + g];
        float clsl = 0.0f;
        for (int c = 0; c < NUM_CLASSES; ++c) {
            float sc = sigf(pb[(size_t)(5 + c) * hw]);
            float t = (c == cls) ? 1.0f : 0.0f;
            float d = sc - t;
            clsl += d * d;
        }
        loss += clsl;
    }
    block_accum(loss, part_slot_base + blockIdx.x);
}

// fixed-order final reduction (bitwise deterministic)
__global__ void k_final(const float* __restrict__ part, float* __restrict__ out) {
    __shared__ float red[256];
    float s = 0.0f;
    for (int i = threadIdx.x; i < NPART; i += 256) s += part[i];
    red[threadIdx.x] = s;
    __syncthreads();
    for (int off = 128; off > 0; off >>= 1) {
        if (threadIdx.x < off) red[threadIdx.x] += red[threadIdx.x + off];
        __syncthreads();
    }
    if (threadIdx.x == 0) out[0] = red[0];
}

// ----------------------------------------------------------------- launcher
extern "C" void kernel_launch(void* const* d_in, const int* in_sizes, int n_in,
                              void* d_out, int out_size, void* d_ws, size_t ws_size,
                              hipStream_t stream) {
    const float* p0  = (const float*)d_in[0];
    const float* p1  = (const float*)d_in[1];
    const float* p2  = (const float*)d_in[2];
    const float* gtb = (const float*)d_in[3];
    const int*   gtc = (const int*)d_in[4];
    const float* anc = (const float*)d_in[5];

    unsigned char* flags = (unsigned char*)d_ws;
    float* rec  = (float*)((char*)d_ws + REC_OFF);
    float* part = (float*)((char*)d_ws + PART_OFF);
    float* out  = (float*)d_out;

    k_zero<<<(FLAG_WORDS + 255) / 256, 256, 0, stream>>>((unsigned int*)flags);
    k_assign<<<(NREC + 255) / 256, 256, 0, stream>>>(gtb, anc, flags, rec);

    k_noobj<<<BLK0, 256, 0, stream>>>(p0, anc + 0, gtb, flags + SB0,
                                      13, NB0, BATCH * NB0, part + SLOT0);
    k_noobj<<<BLK1, 256, 0, stream>>>(p1, anc + 6, gtb, flags + SB1,
                                      26, NB1, BATCH * NB1, part + SLOT1);
    k_noobj<<<BLK2, 256, 0, stream>>>(p2, anc + 12, gtb, flags + SB2,
                                      52, NB2, BATCH * NB2, part + SLOT2);

    k_targets<<<BLKT, 256, 0, stream>>>(p0, p1, p2, gtb, gtc, anc, rec,
                                        part + SLOTT);
    k_final<<<1, 256, 0, stream>>>(part, out);
}